// NT_Xent_39591008534884
// MI455X (gfx1250) — compile-verified
//
#include <hip/hip_runtime.h>
#include <math.h>

typedef __attribute__((ext_vector_type(2))) float v2f;
typedef __attribute__((ext_vector_type(8))) float v8f;

#define BROWS 4096                  // B
#define DDIM  128                   // D
#define NTOT  8192                  // N = 2B
#define NTILE (NTOT / 16)           // 512 column tiles
#define INV_T 14.2857142857142857f  // 1/0.07 == max possible logit (cos<=1)
#define EPSN  1e-8f
#define LROW  132                   // LDS row stride (floats): conflict-free banks

// Map logical row i of z = cat(zz[:,0], zz[:,1]) to flat offset in zz[B,2,D].
__device__ __forceinline__ size_t row_off(int i) {
    return (i < BROWS) ? (size_t)i * (2 * DDIM)
                       : (size_t)(i - BROWS) * (2 * DDIM) + DDIM;
}

// ---------------- Kernel 1: per-row inverse norms (one wave per row) --------
__global__ void ntxent_norms(const float* __restrict__ zz,
                             float* __restrict__ invn) {
    int gwave = (blockIdx.x * blockDim.x + threadIdx.x) >> 5;
    int lane  = threadIdx.x & 31;
    if (gwave >= NTOT) return;
    const float* src = zz + row_off(gwave);
    float ss = 0.f;
    for (int k = lane; k < DDIM; k += 32) { float v = src[k]; ss += v * v; }
    #pragma unroll
    for (int m = 16; m >= 1; m >>= 1) ss += __shfl_xor(ss, m, 32);
    if (lane == 0) invn[gwave] = 1.0f / fmaxf(sqrtf(ss), EPSN);
}

// Issue the 16 async B128 copies that move one 16x128-f32 column tile
// (rows c0..c0+15 of z) into a padded LDS buffer. Tracked by ASYNCcnt.
// lane role: row = lane&15, seg = lane>>4; instr t copies bytes
// [16*seg + 32*t, +16) of the row (same offset applied to LDS dst & global).
__device__ __forceinline__ void issue_tile_async(const float* __restrict__ zz,
                                                 int c0, float* buf, int lane) {
    const int r   = lane & 15;
    const int seg = lane >> 4;
    const unsigned long long ga =
        (unsigned long long)(uintptr_t)(zz + row_off(c0 + r) + seg * 4);
    const unsigned la =
        (unsigned)(uintptr_t)(buf + r * LROW + seg * 4); // low 32 bits = LDS offset
    #pragma unroll
    for (int t = 0; t < 16; ++t) {
        asm volatile("global_load_async_to_lds_b128 %0, %1, off offset:%2"
                     :: "v"(la), "v"(ga), "i"(t * 32) : "memory");
    }
}

// ---------------- Kernel 2: fused sim-GEMM + online logsumexp ---------------
// One wave owns a 16-row tile; sweeps all 512 column tiles with
// V_WMMA_F32_16X16X4_F32 (exact fp32). Column tiles are double-buffered in
// LDS via global_load_async_to_lds_b128 / s_wait_asynccnt (per-wave pipeline,
// no cross-wave barriers). sim is never materialized.
__global__ __launch_bounds__(256) void ntxent_main(const float* __restrict__ zz,
                                                   const float* __restrict__ invn,
                                                   float* __restrict__ rowloss) {
    extern __shared__ float lds[];                // 8 waves * 2 bufs * 16*LROW
    const int lane = threadIdx.x & 31;
    const int wib  = threadIdx.x >> 5;            // wave in block (0..7)
    const int tile = blockIdx.x * 8 + wib;        // row-tile id (0..511)
    const int r0   = tile * 16;
    const int nloc = lane & 15;                   // M index for A, N index for B/C
    const int half = lane >> 4;                   // K-half select

    float* buf0 = lds + (size_t)wib * 2 * (16 * LROW);
    float* buf1 = buf0 + 16 * LROW;

    // ---- Preload & pre-scale A fragments: A[m][k], m = r0+nloc -------------
    const int   arow   = r0 + nloc;
    const float ascale = invn[arow] * INV_T;      // fold 1/T into A
    const float* aptr  = zz + row_off(arow);
    v2f afrag[32];
    #pragma unroll
    for (int j = 0; j < 32; ++j) {                // frag j covers K = 4j..4j+3
        const int k = 4 * j + 2 * half;           // lanes 16-31 take K=2h,2h+1
        v2f a; a.x = aptr[k]; a.y = aptr[k + 1];
        afrag[j] = a * ascale;
    }

    // Rows owned by this lane in the C layout, and their positive partners.
    int rowi[8], part[8];
    #pragma unroll
    for (int v = 0; v < 8; ++v) {
        rowi[v] = r0 + 8 * half + v;
        part[v] = (rowi[v] < BROWS) ? rowi[v] + BROWS : rowi[v] - BROWS;
    }

    float ssum[8], posv[8];
    #pragma unroll
    for (int v = 0; v < 8; ++v) { ssum[v] = 0.f; posv[v] = 0.f; }

    // ---- Prime the async pipeline -----------------------------------------
    issue_tile_async(zz, 0, buf0, lane);

    // ---- Sweep all column tiles -------------------------------------------
    for (int t = 0; t < NTILE; ++t) {
        const int c0  = t * 16;
        float* cur = (t & 1) ? buf1 : buf0;
        float* nxt = (t & 1) ? buf0 : buf1;

        if (t + 1 < NTILE) {
            // nxt's previous readers (iteration t-1) must be fully drained
            // before the async engine overwrites it.
            asm volatile("s_wait_dscnt 0x0" ::: "memory");
            issue_tile_async(zz, c0 + 16, nxt, lane);
            // <=16 outstanding ==> the batch for `cur` has landed in LDS.
            asm volatile("s_wait_asynccnt 0x10" ::: "memory");
        } else {
            asm volatile("s_wait_asynccnt 0x0" ::: "memory");
        }

        v8f c = {0.f, 0.f, 0.f, 0.f, 0.f, 0.f, 0.f, 0.f};
        const float* brow = cur + nloc * LROW;    // this lane's B column data
        #pragma unroll
        for (int j = 0; j < 32; ++j) {
            const int k = 4 * j + 2 * half;
            v2f b; b.x = brow[k]; b.y = brow[k + 1];   // ds_load_b64, bank-clean
            c = __builtin_amdgcn_wmma_f32_16x16x4_f32(
                    false, afrag[j], false, b, (short)0, c, false, false);
        }

        // Column normalization applied once per output: this lane's C column
        // (c0+nloc) is exactly the row it staged, so scale after the WMMAs.
        const int   col    = c0 + nloc;
        const float bscale = invn[col];
        #pragma unroll
        for (int v = 0; v < 8; ++v) {
            const float x = c[v] * bscale;             // sim(rowi[v], col)/T
            posv[v] = (col == part[v]) ? x : posv[v];  // one-hot capture
            const float e = __expf(x - INV_T);         // exact fixed max bound
            ssum[v] += (col == rowi[v]) ? 0.f : e;     // exclude self
        }
    }

    // ---- Reduce across the 16 lanes sharing each row (xor stays in half) --
    #pragma unroll
    for (int v = 0; v < 8; ++v) {
        float s = ssum[v], p = posv[v];
        #pragma unroll
        for (int m = 1; m <= 8; m <<= 1) {
            s += __shfl_xor(s, m, 16);
            p += __shfl_xor(p, m, 16);   // one-hot: sum == the captured value
        }
        ssum[v] = s; posv[v] = p;
    }
    if (nloc == 0) {
        #pragma unroll
        for (int v = 0; v < 8; ++v)
            rowloss[rowi[v]] = (logf(ssum[v]) + INV_T) - posv[v];
    }
}

// ---------------- Kernel 3: deterministic mean reduction --------------------
__global__ void ntxent_reduce(const float* __restrict__ rowloss,
                              float* __restrict__ out) {
    __shared__ float sm[256];
    float s = 0.f;
    for (int i = threadIdx.x; i < NTOT; i += 256) s += rowloss[i];
    sm[threadIdx.x] = s;
    __syncthreads();
    for (int w = 128; w > 0; w >>= 1) {
        if ((int)threadIdx.x < w) sm[threadIdx.x] += sm[threadIdx.x + w];
        __syncthreads();
    }
    if (threadIdx.x == 0) out[0] = sm[0] * (1.0f / (float)NTOT);
}

extern "C" void kernel_launch(void* const* d_in, const int* in_sizes, int n_in,
                              void* d_out, int out_size, void* d_ws, size_t ws_size,
                              hipStream_t stream) {
    (void)in_sizes; (void)n_in; (void)out_size; (void)ws_size;
    const float* zz  = (const float*)d_in[0];
    float* out       = (float*)d_out;
    float* invn      = (float*)d_ws;          // [NTOT] floats
    float* rowloss   = invn + NTOT;           // [NTOT] floats (64 KB total ws)

    // 1) inverse norms: one wave per row -> 8192 waves
    ntxent_norms<<<NTOT / 8, 256, 0, stream>>>(zz, invn);
    // 2) fused GEMM + logsumexp: 512 row tiles, 8 waves/block,
    //    dynamic LDS: 8 waves * 2 buffers * 16*LROW floats
    const size_t shbytes = (size_t)8 * 2 * 16 * LROW * sizeof(float);
    ntxent_main<<<(NTOT / 16) / 8, 256, shbytes, stream>>>(zz, invn, rowloss);
    // 3) deterministic mean
    ntxent_reduce<<<1, 256, 0, stream>>>(rowloss, out);
}